// EigVals_54872502174209
// MI455X (gfx1250) — compile-verified
//
#include <hip/hip_runtime.h>
#include <hip/hip_bf16.h>
#include <math.h>

typedef float        f32x4 __attribute__((ext_vector_type(4)));
typedef unsigned int u32x4 __attribute__((ext_vector_type(4)));
typedef int          i32x8 __attribute__((ext_vector_type(8)));
typedef int          i32x4 __attribute__((ext_vector_type(4)));

#define TWO_PI_3 2.0943951023931953f

#define BLOCK_T   128                 // 4 wave32s
#define VOX_PT    4                   // voxels per thread (b128 granularity)
#define CHUNK_VOX (BLOCK_T * VOX_PT)  // 512 voxels per chunk

// Closed-form symmetric 3x3 eigensolver (trace-shift + acos method).
__device__ __forceinline__ void eig3(float a00, float a01, float a02,
                                     float a11, float a12, float a22,
                                     float& lo_out, float& mid_out, float& hi_out)
{
    const float q   = (a00 + a11 + a22) * (1.0f / 3.0f);
    const float b00 = a00 - q, b11 = a11 - q, b22 = a22 - q;
    const float p2  = b00 * b00 + b11 * b11 + b22 * b22
                    + 2.0f * (a01 * a01 + a02 * a02 + a12 * a12);
#if defined(__HIP_DEVICE_COMPILE__)
    const float p    = __builtin_amdgcn_sqrtf(p2 * (1.0f / 6.0f));
    const float pinv = (p2 > 1e-30f) ? __builtin_amdgcn_rcpf(p) : 0.0f;
#else
    const float p    = sqrtf(p2 * (1.0f / 6.0f));
    const float pinv = (p2 > 1e-30f) ? 1.0f / p : 0.0f;
#endif
    // B = (A - q I) / p ; r = det(B)/2 clamped to [-1,1]
    const float n00 = b00 * pinv, n11 = b11 * pinv, n22 = b22 * pinv;
    const float n01 = a01 * pinv, n02 = a02 * pinv, n12 = a12 * pinv;
    const float halfdet = 0.5f * ( n00 * (n11 * n22 - n12 * n12)
                                 - n01 * (n01 * n22 - n12 * n02)
                                 + n02 * (n01 * n12 - n11 * n02) );
    const float r   = fminf(fmaxf(halfdet, -1.0f), 1.0f);
    const float phi = acosf(r) * (1.0f / 3.0f);          // [0, pi/3]
    const float two_p = 2.0f * p;
    const float e_hi  = q + two_p * cosf(phi);           // analytic max
    const float e_lo  = q + two_p * cosf(phi + TWO_PI_3);// analytic min
    float lo  = fminf(e_lo, e_hi);
    float hi  = fmaxf(e_lo, e_hi);
    float mid = 3.0f * q - lo - hi;
    lo_out  = fminf(lo, mid);
    hi_out  = fmaxf(hi, mid);
    mid_out = 3.0f * q - lo_out - hi_out;
}

#if defined(__HIP_DEVICE_COMPILE__) && \
    __has_builtin(__builtin_amdgcn_tensor_load_to_lds) && \
    __has_builtin(__builtin_amdgcn_s_wait_tensorcnt)
#define EIG_USE_TDM 1
#else
#define EIG_USE_TDM 0
#endif

#if EIG_USE_TDM
// Build a 2D D# (9 rows x CHUNK_VOX f32, row stride = channel stride) and
// issue TENSOR_LOAD_TO_LDS. Wave-uniform arguments; EXEC is ignored by TDM.
__device__ __forceinline__ void tdm_issue_chunk(const float* x, int nPerBatch,
                                                int chunksPerBatch, int c,
                                                const f32x4* ldsDst)
{
    const int b  = c / chunksPerBatch;
    const long long n0 = (long long)(c - b * chunksPerBatch) * CHUNK_VOX;
    const unsigned long long gaddr =
        (unsigned long long)x
        + ((unsigned long long)b * 9ull * (unsigned long long)nPerBatch
           + (unsigned long long)n0) * 4ull;
    // Generic LDS pointer = {aperture_hi32, lds_offset32}: low 32 bits = DS offset.
    const unsigned ldsOff = (unsigned)(unsigned long long)ldsDst;

    u32x4 g0;
    g0.x = 1u;                                   // count=1 (valid descriptor)
    g0.y = ldsOff;                               // lds_addr (bytes)
    g0.z = (unsigned)(gaddr & 0xFFFFFFFFull);    // global_addr[31:0]
    g0.w = (unsigned)(gaddr >> 32) | 0x80000000u;// global_addr[56:32] | type=2

    i32x8 g1;
    g1[0] = 0x20000;                 // workgroup_mask=0, data_size=2 (4B), no flags
    g1[1] = 0;                       // atomic_barrier_addr=0, tensor_dim0[15:0]=0
    g1[2] = 0x00100010;              // tensor_dim0=0x100000 (hi16), tensor_dim1=16
    g1[3] = (CHUNK_VOX << 16);       // tensor_dim1[31:16]=0, tile_dim0=CHUNK_VOX
    g1[4] = 9;                       // tile_dim1=9 rows, tile_dim2=0
    g1[5] = nPerBatch;               // tensor_dim0_stride[31:0] (elements)
    g1[6] = 0;                       // stride0[47:32]=0, stride1[15:0]=0
    g1[7] = 0;                       // stride1[47:16]=0
    i32x4 gz4 = {0, 0, 0, 0};        // groups 2/3 unused (2D tensor)
    i32x8 gz8 = {0, 0, 0, 0, 0, 0, 0, 0};
    // This toolchain declares the 6-arg form (extra int32x8 group, zero-filled).
    __builtin_amdgcn_tensor_load_to_lds(g0, g1, gz4, gz4, gz8, 0);
}
#endif

__global__ __launch_bounds__(BLOCK_T) void EigVals3x3_kernel(
    const float* __restrict__ x,   // (B, 9, N) f32, channel stride = nPerBatch
    float* __restrict__ out,       // (B*N, 3) f32, ascending eigenvalues
    int nPerBatch,                 // D*H*W, multiple of CHUNK_VOX
    int nChunks)                   // total voxels / CHUNK_VOX
{
    const int t  = (int)threadIdx.x;
    const int c0 = (int)blockIdx.x;
    const int gstride = (int)gridDim.x;
    if (c0 >= nChunks) return;

#if EIG_USE_TDM
    // ---- TDM double-buffered pipeline: global -> LDS via Tensor Data Mover ----
    __shared__ f32x4 sbuf[2][9 * BLOCK_T];       // 2 x 18KB tiles
    const int chunksPerBatch = nPerBatch / CHUNK_VOX;
    const int iters = (nChunks - c0 + gstride - 1) / gstride;

    if (t < 32)  // wave 0 drives the TDM
        tdm_issue_chunk(x, nPerBatch, chunksPerBatch, c0, &sbuf[0][0]);

    for (int i = 0; i < iters; ++i) {
        const int c  = c0 + i * gstride;
        const int bi = i & 1;
        if (t < 32) {
            if (i + 1 < iters) {
                tdm_issue_chunk(x, nPerBatch, chunksPerBatch, c + gstride,
                                &sbuf[1 - bi][0]);
                __builtin_amdgcn_s_wait_tensorcnt(1);  // chunk i resident (in-order)
            } else {
                __builtin_amdgcn_s_wait_tensorcnt(0);
            }
        }
        __syncthreads();                                // tile visible to all waves

        f32x4 ch[9];
#pragma unroll
        for (int k = 0; k < 9; ++k)                     // ds_load_b128 per channel
            ch[k] = sbuf[bi][k * BLOCK_T + t];

        const long long v = (long long)c * CHUNK_VOX + (long long)t * VOX_PT;
        float res[12];
#pragma unroll
        for (int j = 0; j < VOX_PT; ++j) {
            eig3(ch[0][j], 0.5f * (ch[1][j] + ch[3][j]), 0.5f * (ch[2][j] + ch[6][j]),
                 ch[4][j], 0.5f * (ch[5][j] + ch[7][j]), ch[8][j],
                 res[j * 3 + 0], res[j * 3 + 1], res[j * 3 + 2]);
        }
        f32x4* o = (f32x4*)(out + v * 3);
        f32x4 o0 = { res[0], res[1], res[2],  res[3]  };
        f32x4 o1 = { res[4], res[5], res[6],  res[7]  };
        f32x4 o2 = { res[8], res[9], res[10], res[11] };
        __builtin_nontemporal_store(o0, o + 0);
        __builtin_nontemporal_store(o1, o + 1);
        __builtin_nontemporal_store(o2, o + 2);

        __syncthreads();                                // buffer reusable
    }
#else
    // ---- Fallback: direct NT b128 streaming loads (+ gfx1250 prefetch) ----
    const long long cs = (long long)nPerBatch;
    for (int c = c0; c < nChunks; c += gstride) {
        const long long v = (long long)c * CHUNK_VOX + (long long)t * VOX_PT;
        const int b = (int)(v / (long long)nPerBatch);
        const long long n = v - (long long)b * nPerBatch;
        const float* base = x + (long long)b * 9ll * cs + n;

        if (c + gstride < nChunks) {
            const char* pf = (const char*)(base + (long long)gstride * CHUNK_VOX);
#pragma unroll
            for (int k = 0; k < 9; ++k)
                __builtin_prefetch(pf + (long long)k * cs * (long long)sizeof(float), 0, 1);
        }
        f32x4 ch[9];
#pragma unroll
        for (int k = 0; k < 9; ++k)
            ch[k] = __builtin_nontemporal_load((const f32x4*)(base + (long long)k * cs));

        float res[12];
#pragma unroll
        for (int j = 0; j < VOX_PT; ++j) {
            eig3(ch[0][j], 0.5f * (ch[1][j] + ch[3][j]), 0.5f * (ch[2][j] + ch[6][j]),
                 ch[4][j], 0.5f * (ch[5][j] + ch[7][j]), ch[8][j],
                 res[j * 3 + 0], res[j * 3 + 1], res[j * 3 + 2]);
        }
        f32x4* o = (f32x4*)(out + v * 3);
        f32x4 o0 = { res[0], res[1], res[2],  res[3]  };
        f32x4 o1 = { res[4], res[5], res[6],  res[7]  };
        f32x4 o2 = { res[8], res[9], res[10], res[11] };
        __builtin_nontemporal_store(o0, o + 0);
        __builtin_nontemporal_store(o1, o + 1);
        __builtin_nontemporal_store(o2, o + 2);
    }
#endif
}

extern "C" void kernel_launch(void* const* d_in, const int* in_sizes, int n_in,
                              void* d_out, int out_size, void* d_ws, size_t ws_size,
                              hipStream_t stream) {
    (void)n_in; (void)d_ws; (void)ws_size; (void)out_size;

    const float* x = (const float*)d_in[0];
    float* out = (float*)d_out;

    // Reference fixes the volume shape: (B, 9, 64, 192, 192)
    const int nPerBatch = 64 * 192 * 192;            // 2,359,296 voxels / batch
    const int totalVox  = in_sizes[0] / 9;           // B * nPerBatch
    const int nChunks   = totalVox / CHUNK_VOX;      // 9216 chunks of 512 voxels

    // 2304 blocks x 128 threads = 9216 wave32s; 4 chunks per block so the
    // TDM double-buffer overlaps chunk i+1's DMA with chunk i's compute.
    int grid = 2304;
    if (grid > nChunks) grid = nChunks;
    if (grid < 1) grid = 1;

    EigVals3x3_kernel<<<dim3(grid), dim3(BLOCK_T), 0, stream>>>(x, out, nPerBatch, nChunks);
}